// LocalLuongAttention_76106820485228
// MI455X (gfx1250) — compile-verified
//
#include <hip/hip_runtime.h>
#include <math.h>

// Problem dims (fixed by the reference)
#define BATCH 32
#define SEQ   4096
#define DIM   1024
#define OUTD  1024
#define SCHUNK 16                          // split SEQ into 16 chunks
#define ROWS_PER_CHUNK (SEQ / SCHUNK)      // 256 rows per block
#define TILE_ROWS 32                       // rows staged in LDS per tile
#define TILES_PER_CHUNK (ROWS_PER_CHUNK / TILE_ROWS) // 8
#define INV_2SIG2 (1.0f / 128.0f)          // 1/(2*8*8), STDDEV=8

typedef __attribute__((ext_vector_type(2))) float v2f;
typedef __attribute__((ext_vector_type(8))) float v8f;

// ---------------------------------------------------------------------------
// Kernel 1: fused flash-style score + online softmax + Gaussian-pos-weighted
// accumulation. Reads X exactly once (512 MB total -> ~22us at 23.3 TB/s).
// grid = (SCHUNK, BATCH), block = 256 threads (8 waves), dyn LDS ~128 KB
// (legal on CDNA5: 320 KB/WGP; 2 blocks co-resident per WGP for overlap).
// ---------------------------------------------------------------------------
__global__ void flash_partial_kernel(const float* __restrict__ X,   // [B][S][D]
                                     const float* __restrict__ T,   // [B][D]
                                     const int*   __restrict__ pos, // [B]
                                     float* __restrict__ pacc,      // [B][SCHUNK][D]
                                     float* __restrict__ pm,        // [B][SCHUNK]
                                     float* __restrict__ pz)        // [B][SCHUNK]
{
    extern __shared__ float smem[];            // TILE_ROWS*DIM + 2*TILE_ROWS + 1
    float* tile  = smem;                       // [TILE_ROWS][DIM] staged X
    float* ssc   = smem + TILE_ROWS * DIM;     // [TILE_ROWS] raw scores
    float* scoef = ssc + TILE_ROWS;            // [TILE_ROWS] exp coefs (w/ pos_w)
    float* szsum = scoef + TILE_ROWS;          // [1] tile sum of exp

    const int b    = blockIdx.y;
    const int sc   = blockIdx.x;
    const int tid  = threadIdx.x;
    const int wave = tid >> 5;
    const int lane = tid & 31;
    const int s0   = sc * ROWS_PER_CHUNK;

    const float* xb = X + ((size_t)b * SEQ + s0) * DIM;
    const float* tb = T + (size_t)b * DIM;
    const float  p  = (float)pos[b];

    const int d0 = tid * 4;                    // accumulate: 4 contiguous d / thread
    float m = -INFINITY, Z = 0.0f;
    float4 acc = make_float4(0.f, 0.f, 0.f, 0.f);

    for (int t = 0; t < TILES_PER_CHUNK; ++t) {
        const int rbase = t * TILE_ROWS;

        // ---- Phase 1: each wave computes dots for 4 rows, staging X into LDS
        for (int rr = 0; rr < 4; ++rr) {
            const int row = wave + 8 * rr;                 // 0..31
            const float* xr = xb + (size_t)(rbase + row) * DIM;
            float dot = 0.f;
            #pragma unroll
            for (int i = 0; i < 8; ++i) {
                const int d = lane * 4 + i * 128;
                float4 xv = *(const float4*)(xr + d);
                float4 tv = *(const float4*)(tb + d);
                *(float4*)(tile + row * DIM + d) = xv;
                dot += xv.x * tv.x + xv.y * tv.y + xv.z * tv.z + xv.w * tv.w;
            }
            #pragma unroll
            for (int off = 16; off > 0; off >>= 1)
                dot += __shfl_xor(dot, off, 32);
            if (lane == 0) ssc[row] = dot;
        }
        __syncthreads();

        // ---- Phase 2a: per-wave (redundant across waves, bitwise identical):
        // tile max via lane shuffle, each lane handles row==lane's coefficient.
        const float sv = ssc[lane];                        // one LDS read / lane
        float tmax = sv;
        #pragma unroll
        for (int off = 16; off > 0; off >>= 1)
            tmax = fmaxf(tmax, __shfl_xor(tmax, off, 32));
        const float mnew  = fmaxf(m, tmax);
        const float scale = expf(m - mnew);                // expf(-inf)=0 first tile
        const float e     = expf(sv - mnew);               // softmax numerator (row=lane)
        const float rel   = (float)(s0 + rbase + lane) - p;
        const float coef  = e * expf(-(rel * rel) * INV_2SIG2); // Gaussian decay
        float zs = e;                                      // Z excludes pos_w (ref!)
        #pragma unroll
        for (int off = 16; off > 0; off >>= 1)
            zs += __shfl_xor(zs, off, 32);
        if (wave == 0) {                                   // stage coefs once
            scoef[lane] = coef;
            if (lane == 0) szsum[0] = zs;
        }
        Z = Z * scale + zs;
        acc.x *= scale; acc.y *= scale; acc.z *= scale; acc.w *= scale;
        m = mnew;
        __syncthreads();

        // ---- Phase 2b: accumulate weighted sum from LDS (1 coef broadcast +
        // 1 float4 LDS read + 4 FMA per row per thread)
        #pragma unroll 8
        for (int r = 0; r < TILE_ROWS; ++r) {
            const float c = scoef[r];
            float4 xv = *(const float4*)(tile + r * DIM + d0);
            acc.x += c * xv.x; acc.y += c * xv.y;
            acc.z += c * xv.z; acc.w += c * xv.w;
        }
        __syncthreads();
    }

    *(float4*)(pacc + ((size_t)b * SCHUNK + sc) * DIM + d0) = acc;
    if (tid == 0) { pm[b * SCHUNK + sc] = m; pz[b * SCHUNK + sc] = Z; }
}

// ---------------------------------------------------------------------------
// Kernel 2: merge split-softmax partials; build combined = [target | weighted]
// grid = BATCH, block = 256 (4 d's per thread)
// ---------------------------------------------------------------------------
__global__ void combine_kernel(const float* __restrict__ pacc,
                               const float* __restrict__ pm,
                               const float* __restrict__ pz,
                               const float* __restrict__ T,
                               float* __restrict__ comb)            // [B][2*DIM]
{
    const int b = blockIdx.x;
    const int tid = threadIdx.x;
    const int d0 = tid * 4;

    float M = -INFINITY;
    #pragma unroll
    for (int i = 0; i < SCHUNK; ++i) M = fmaxf(M, pm[b * SCHUNK + i]);
    float e[SCHUNK];
    float Z = 0.f;
    #pragma unroll
    for (int i = 0; i < SCHUNK; ++i) {
        e[i] = expf(pm[b * SCHUNK + i] - M);
        Z += pz[b * SCHUNK + i] * e[i];
    }
    float4 acc = make_float4(0.f, 0.f, 0.f, 0.f);
    #pragma unroll
    for (int i = 0; i < SCHUNK; ++i) {
        float4 pv = *(const float4*)(pacc + ((size_t)b * SCHUNK + i) * DIM + d0);
        acc.x += pv.x * e[i]; acc.y += pv.y * e[i];
        acc.z += pv.z * e[i]; acc.w += pv.w * e[i];
    }
    const float inv = 1.0f / Z;
    acc.x *= inv; acc.y *= inv; acc.z *= inv; acc.w *= inv;

    float4 tv = *(const float4*)(T + (size_t)b * DIM + d0);
    *(float4*)(comb + (size_t)b * 2 * DIM + d0)        = tv;
    *(float4*)(comb + (size_t)b * 2 * DIM + DIM + d0)  = acc;
}

// ---------------------------------------------------------------------------
// Kernel 3: out = tanh(combined[32x2048] @ W[2048x1024]) via fp32 WMMA 16x16x4.
// 128 output tiles of 16x16; one wave per tile; grid = 32 blocks x 4 waves.
// A layout (16x4 f32): lanes 0-15 hold K=0,1; lanes 16-31 hold K=2,3.
// C/D layout: VGPR r -> row r (lanes 0-15) / row r+8 (lanes 16-31), col = lane&15.
// ---------------------------------------------------------------------------
__global__ void gemm_tanh_wmma_kernel(const float* __restrict__ A,   // [32][2048]
                                      const float* __restrict__ W,   // [2048][1024]
                                      float* __restrict__ out)       // [32][1024]
{
    const int wave = threadIdx.x >> 5;
    const int lane = threadIdx.x & 31;
    const int tile = blockIdx.x * 4 + wave;   // 0..127
    const int mt = tile >> 6;                 // 0..1  (M tiles)
    const int nt = tile & 63;                 // 0..63 (N tiles)
    const int m0 = mt * 16, n0 = nt * 16;
    const int lh  = lane >> 4;                // lane-half selector
    const int l15 = lane & 15;

    v8f c = {};
    const float* arow = A + (size_t)(m0 + l15) * (2 * DIM);
    for (int k = 0; k < 2 * DIM; k += 4) {
        const int kk = k + lh * 2;
        v2f a = *(const v2f*)(arow + kk);                 // A[m0+l15][kk..kk+1]
        const float* wp = W + (size_t)kk * OUTD + n0 + l15;
        v2f bvec; bvec.x = wp[0]; bvec.y = wp[OUTD];      // B[kk..kk+1][n0+l15]
        // (neg_a, A, neg_b, B, c_mod, C, reuse_a, reuse_b)
        c = __builtin_amdgcn_wmma_f32_16x16x4_f32(false, a, false, bvec,
                                                  (short)0, c, false, false);
    }
    #pragma unroll
    for (int r = 0; r < 8; ++r) {
        const int row = m0 + r + lh * 8;
        const int col = n0 + l15;
        out[(size_t)row * OUTD + col] = tanhf(c[r]);
    }
}

// ---------------------------------------------------------------------------
extern "C" void kernel_launch(void* const* d_in, const int* in_sizes, int n_in,
                              void* d_out, int out_size, void* d_ws, size_t ws_size,
                              hipStream_t stream) {
    const float* X   = (const float*)d_in[0];  // [32,4096,1024]
    const float* T   = (const float*)d_in[1];  // [32,1024]
    const int*   pos = (const int*)d_in[2];    // [32]
    const float* W   = (const float*)d_in[3];  // [2048,1024]
    float* out = (float*)d_out;                // [32,1024]

    char* ws = (char*)d_ws;
    const size_t pacc_bytes = (size_t)BATCH * SCHUNK * DIM * sizeof(float); // 2 MB
    float* pacc = (float*)ws;
    float* pm   = (float*)(ws + pacc_bytes);                 // 2 KB
    float* pz   = pm + BATCH * SCHUNK;                       // 2 KB
    float* comb = (float*)(ws + pacc_bytes + 8192);          // 256 KB, 16B aligned

    const size_t smem =
        (size_t)(TILE_ROWS * DIM + 2 * TILE_ROWS + 1) * sizeof(float); // ~128 KB

    flash_partial_kernel<<<dim3(SCHUNK, BATCH), 256, smem, stream>>>(X, T, pos,
                                                                     pacc, pm, pz);
    combine_kernel<<<BATCH, 256, 0, stream>>>(pacc, pm, pz, T, comb);
    gemm_tanh_wmma_kernel<<<BATCH, 128, 0, stream>>>(comb, W, out);
}